// _AvgPoolSSIM_8486855377077
// MI455X (gfx1250) — compile-verified
//
#include <hip/hip_runtime.h>
#include <stdint.h>

#ifndef __has_builtin
#define __has_builtin(x) 0
#endif

#define BATCH 32
#define CHN   3
#define HH    512
#define WW    512
#define ROWS  64
#define STRIPS (HH / ROWS)          // 8
#define NTHREADS 256
#define BUFW  520                   // 4 pad + 512 + 4 pad
#define NITER (ROWS + 6)            // 70
#define NGROUP (NITER / 7)          // 10

#define SSIM_C1 0.0001f             // 0.01^2
#define SSIM_C2 0.0009f             // 0.03^2
#define INV_WIN (1.0f / 49.0f)

typedef __attribute__((ext_vector_type(2))) float v2f;
typedef __attribute__((ext_vector_type(4))) float v4f;
typedef __attribute__((ext_vector_type(8))) float v8f;
typedef int v4i __attribute__((vector_size(16)));   // matches builtin prototype

#define AS1 __attribute__((address_space(1)))
#define AS3 __attribute__((address_space(3)))

#if defined(__AMDGCN__) && __has_builtin(__builtin_amdgcn_global_load_async_to_lds_b128) && __has_builtin(__builtin_amdgcn_s_wait_asynccnt)
#define USE_ASYNC_LDS 1
#else
#define USE_ASYNC_LDS 0
#endif

__device__ __forceinline__ void stage16(const float* gsrc, float* ldst) {
#if USE_ASYNC_LDS
  __builtin_amdgcn_global_load_async_to_lds_b128(
      (AS1 v4i*)gsrc,
      (AS3 v4i*)ldst,
      0, 0);
#else
  *(v4f*)ldst = *(const v4f*)gsrc;
#endif
}

__device__ __forceinline__ void wait_stage() {
#if USE_ASYNC_LDS
  __builtin_amdgcn_s_wait_asynccnt(0);
#endif
}

__global__ __launch_bounds__(NTHREADS)
void ssim_strip_kernel(const float* __restrict__ x, const float* __restrict__ y,
                       float* __restrict__ partials) {
  __shared__ alignas(16) float rowbuf[2][2][BUFW];   // [buf][x/y][padded row]
  __shared__ float wacc[NTHREADS / 32];

  const int tid   = threadIdx.x;
  const int strip = blockIdx.x;
  const int ch    = blockIdx.y;
  const int b     = blockIdx.z;
  const int r0    = strip * ROWS;
  const int rbase = r0 - 3;

  const size_t img = (size_t)(b * CHN + ch) * (size_t)(HH * WW);
  const float* xim = x + img;
  const float* yim = y + img;

  // row staging: threads 0..127 stage x (4 cols each), 128..255 stage y
  const int arr = tid >> 7;
  const int c4  = (tid & 127) * 4;
  const float* gim = arr ? yim : xim;

  // zero the horizontal padding columns of both row buffers (stay zero forever)
  if (tid < 32) {
    int nb = tid >> 4, rem = tid & 15, a = rem >> 3, k = rem & 7;
    int idx = (k < 4) ? k : 512 + k;
    rowbuf[nb][a][idx] = 0.0f;
  }

  // stage first input row (rbase) into buffer 0 (zeros if above the image)
  {
    float* dst = &rowbuf[0][arr][4 + c4];
    if (rbase >= 0) {
      stage16(gim + (size_t)rbase * WW + c4, dst);
    } else {
      v4f z = {0.f, 0.f, 0.f, 0.f};
      *(v4f*)dst = z;
    }
  }

  const int w0 = 2 * tid + 1;   // window start index in padded row buffer

  float vs[5][2];               // running vertical sums (5 quantities, 2 cols)
  float ring[5][7][2];          // 7-row ring of horizontal sums
#pragma unroll
  for (int q = 0; q < 5; ++q)
#pragma unroll
    for (int j = 0; j < 2; ++j) {
      vs[q][j] = 0.f;
#pragma unroll
      for (int s = 0; s < 7; ++s) ring[q][s][j] = 0.f;
    }
  float acc = 0.f;

#pragma unroll 1
  for (int g = 0; g < NGROUP; ++g) {
#pragma unroll
    for (int s = 0; s < 7; ++s) {           // fully unrolled: ring slot static
      const int i = g * 7 + s;
      wait_stage();
      __syncthreads();

      // stage next row into the other buffer (overlaps with compute below)
      if (i + 1 < NITER) {
        const int rn = rbase + i + 1;
        float* dst = &rowbuf[(i + 1) & 1][arr][4 + c4];
        if (rn >= 0 && rn < HH) {
          stage16(gim + (size_t)rn * WW + c4, dst);
        } else {
          v4f z = {0.f, 0.f, 0.f, 0.f};
          *(v4f*)dst = z;
        }
      }

      const float* xb = &rowbuf[i & 1][0][0];
      const float* yb = &rowbuf[i & 1][1][0];

      // 8-wide window (cols c0-3 .. c0+4) for this thread's two columns
      float xv[8], yv[8];
      xv[0] = xb[w0];
      { v2f t = *(const v2f*)&xb[w0 + 1]; xv[1] = t.x; xv[2] = t.y; }
      { v2f t = *(const v2f*)&xb[w0 + 3]; xv[3] = t.x; xv[4] = t.y; }
      { v2f t = *(const v2f*)&xb[w0 + 5]; xv[5] = t.x; xv[6] = t.y; }
      xv[7] = xb[w0 + 7];
      yv[0] = yb[w0];
      { v2f t = *(const v2f*)&yb[w0 + 1]; yv[1] = t.x; yv[2] = t.y; }
      { v2f t = *(const v2f*)&yb[w0 + 3]; yv[3] = t.x; yv[4] = t.y; }
      { v2f t = *(const v2f*)&yb[w0 + 5]; yv[5] = t.x; yv[6] = t.y; }
      yv[7] = yb[w0 + 7];

      // horizontal 7-sums for both columns (second via slide)
      float h[5][2];
      float sx = 0.f, sy = 0.f, sxx = 0.f, syy = 0.f, sxy = 0.f;
#pragma unroll
      for (int k = 0; k < 7; ++k) {
        sx += xv[k];
        sy += yv[k];
        sxx = fmaf(xv[k], xv[k], sxx);
        syy = fmaf(yv[k], yv[k], syy);
        sxy = fmaf(xv[k], yv[k], sxy);
      }
      h[0][0] = sx;  h[0][1] = sx  - xv[0]       + xv[7];
      h[1][0] = sy;  h[1][1] = sy  - yv[0]       + yv[7];
      h[2][0] = sxx; h[2][1] = sxx - xv[0]*xv[0] + xv[7]*xv[7];
      h[3][0] = syy; h[3][1] = syy - yv[0]*yv[0] + yv[7]*yv[7];
      h[4][0] = sxy; h[4][1] = sxy - xv[0]*yv[0] + xv[7]*yv[7];

      // running vertical sums with static ring slot s
#pragma unroll
      for (int q = 0; q < 5; ++q)
#pragma unroll
        for (int j = 0; j < 2; ++j) {
          vs[q][j] += h[q][j] - ring[q][s][j];
          ring[q][s][j] = h[q][j];
        }

      if ((g > 0) || (s == 6)) {            // i >= 6: emit output row rbase+i-3
#pragma unroll
        for (int j = 0; j < 2; ++j) {
          float mu_x = vs[0][j] * INV_WIN;
          float mu_y = vs[1][j] * INV_WIN;
          float ex2  = vs[2][j] * INV_WIN;
          float ey2  = vs[3][j] * INV_WIN;
          float exy  = vs[4][j] * INV_WIN;
          float mxx = mu_x * mu_x, myy = mu_y * mu_y, mxy = mu_x * mu_y;
          float sigx = ex2 - mxx, sigy = ey2 - myy, sigxy = exy - mxy;
          float num = (2.f * mxy + SSIM_C1) * (2.f * sigxy + SSIM_C2);
          float den = (mxx + myy + SSIM_C1) * (sigx + sigy + SSIM_C2);
          acc += num / (den + 1e-12f);
        }
      }
    }
  }

  // --- workgroup reduction -------------------------------------------------
  // Wave-level sum via WMMA: D = ones(16x4) x B(4x16) + 0. With A all-ones,
  // D[i][j] = column-sum of B, so summing the 16 column sums equals the sum of
  // all 64 B elements irrespective of B's lane layout (layout-proof).
  float wred;
#if __has_builtin(__builtin_amdgcn_wmma_f32_16x16x4_f32)
  {
    v2f av = {1.0f, 1.0f};
    v2f bv = {acc, 0.0f};
    v8f cz = {0.f, 0.f, 0.f, 0.f, 0.f, 0.f, 0.f, 0.f};
    v8f d = __builtin_amdgcn_wmma_f32_16x16x4_f32(
        false, av, false, bv, (short)0, cz, false, false);
    float t = d[0];                 // column sum for this lane's column
    t += __shfl_xor(t, 1, 32);      // sum the 16 distinct column sums
    t += __shfl_xor(t, 2, 32);      // (each 16-lane half holds all 16)
    t += __shfl_xor(t, 4, 32);
    t += __shfl_xor(t, 8, 32);
    wred = t;
  }
#else
  {
    float t = acc;
    t += __shfl_xor(t, 16, 32);
    t += __shfl_xor(t, 8, 32);
    t += __shfl_xor(t, 4, 32);
    t += __shfl_xor(t, 2, 32);
    t += __shfl_xor(t, 1, 32);
    wred = t;
  }
#endif
  const int lane = tid & 31, wave = tid >> 5;
  if (lane == 0) wacc[wave] = wred;
  __syncthreads();
  if (tid == 0) {
    float t = 0.f;
#pragma unroll
    for (int k = 0; k < NTHREADS / 32; ++k) t += wacc[k];
    partials[(b * CHN + ch) * STRIPS + strip] = t;
  }
}

__global__ __launch_bounds__(32)
void ssim_finalize_kernel(const float* __restrict__ partials, float* __restrict__ out) {
  const int b = blockIdx.x;
  const int t = threadIdx.x;   // 32 threads
  float v = 0.f;
  if (t < CHN * STRIPS) v = partials[b * (CHN * STRIPS) + t];
  v += __shfl_xor(v, 16, 32);
  v += __shfl_xor(v, 8, 32);
  v += __shfl_xor(v, 4, 32);
  v += __shfl_xor(v, 2, 32);
  v += __shfl_xor(v, 1, 32);
  if (t == 0) out[b] = v * (1.0f / (float)(CHN * HH * WW));
}

extern "C" void kernel_launch(void* const* d_in, const int* in_sizes, int n_in,
                              void* d_out, int out_size, void* d_ws, size_t ws_size,
                              hipStream_t stream) {
  (void)in_sizes; (void)n_in; (void)out_size; (void)ws_size;
  const float* x = (const float*)d_in[0];
  const float* y = (const float*)d_in[1];
  float* out      = (float*)d_out;
  float* partials = (float*)d_ws;   // BATCH*CHN*STRIPS = 768 floats

  dim3 grid(STRIPS, CHN, BATCH);
  ssim_strip_kernel<<<grid, NTHREADS, 0, stream>>>(x, y, partials);
  ssim_finalize_kernel<<<BATCH, 32, 0, stream>>>(partials, out);
}